// BAP_36447092474526
// MI455X (gfx1250) — compile-verified
//
#include <hip/hip_runtime.h>

typedef __attribute__((ext_vector_type(2))) float v2f;
typedef __attribute__((ext_vector_type(8))) float v8f;

// Problem constants (from reference): B=16, C=1024, H=W=32 -> HW=1024, M=32
#define NB   16
#define NC   1024
#define NHW  1024
#define NM   32

#define FM_ELEMS  (NB * NM * NC)     // 524288 feature_matrix floats (output part 1)
#define ATT_ELEMS (NB * NM * NHW)    // 524288 attention floats     (output part 2)

// ---------------------------------------------------------------------------
// Kernel 1: attentions = sigmoid(raw), vectorized float4
// ---------------------------------------------------------------------------
__global__ __launch_bounds__(256) void bap_sigmoid_kernel(
    const float* __restrict__ raw, float* __restrict__ att) {
  int i = blockIdx.x * blockDim.x + threadIdx.x;   // one float4 per thread
  float4 v = ((const float4*)raw)[i];
  float4 o;
  o.x = 1.0f / (1.0f + __expf(-v.x));
  o.y = 1.0f / (1.0f + __expf(-v.y));
  o.z = 1.0f / (1.0f + __expf(-v.z));
  o.w = 1.0f / (1.0f + __expf(-v.w));
  ((float4*)att)[i] = o;
}

// ---------------------------------------------------------------------------
// Kernel 2: feature_matrix[b,m,c] = (1/1024) * sum_k att[b,m,k]*features[b,c,k]
// One wave computes a 32x16 (MxC) output slab as TWO 16x16 WMMA tiles sharing
// every B-fragment load, so features stream from memory exactly once.
// Block = 8 waves = (batch b, 128-wide C chunk); all 32 att rows in LDS.
// ---------------------------------------------------------------------------
#define KSTAGE     256
#define LDS_STRIDE 260   // 260 % 64 == 4 -> conflict-free ds_load_b64 pattern

__global__ __launch_bounds__(256) void bap_gemm_wmma_kernel(
    const float* __restrict__ features,   // [B, C, HW]
    const float* __restrict__ att,        // [B, M, HW] (post-sigmoid)
    float* __restrict__ out) {            // [B, M, C]
  __shared__ float lds[NM * LDS_STRIDE];  // 32 rows x 260 = 33.3 KB

  const int blk   = blockIdx.x;     // 128 blocks
  const int b     = blk >> 3;       // 16 batches
  const int chunk = blk & 7;        // which 128 columns of C

  const int lane = threadIdx.x & 31;
  const int wave = threadIdx.x >> 5;
  const int half = lane >> 4;       // 0: K pair {0,1}; 1: K pair {2,3}
  const int l    = lane & 15;       // M row within tile / C col within tile

  const int c_global = chunk * 128 + wave * 16 + l;
  const float* fbase = features + ((size_t)b * NC + c_global) * NHW + half * 2;
  const float* abase = att + (size_t)b * NM * NHW;   // all 32 M rows

  v8f acc0 = (v8f)(0.0f);   // M rows 0..15
  v8f acc1 = (v8f)(0.0f);   // M rows 16..31

  for (int kt = 0; kt < NHW; kt += KSTAGE) {
    __syncthreads();  // previous stage fully consumed
    // Cooperative fill: 32 rows x 256 K = 2048 float4 loads across 256 threads
    #pragma unroll
    for (int t = 0; t < 8; ++t) {
      int idx = (threadIdx.x + t * 256) * 4;     // [0, 8192)
      int row = idx >> 8;                         // /256
      int col = idx & (KSTAGE - 1);
      float4 v = *(const float4*)(abase + row * NHW + kt + col);
      *(float4*)(&lds[row * LDS_STRIDE + col]) = v;
    }
    __syncthreads();

    const float* a0ptr = &lds[l * LDS_STRIDE + half * 2];                  // rows 0..15
    const float* a1ptr = &lds[(16 + l) * LDS_STRIDE + half * 2];           // rows 16..31
    const float* fptr  = fbase + kt;

    // Keep the HBM stream ahead of the WMMA pipe (global_prefetch_b8)
    if (kt + KSTAGE < NHW) __builtin_prefetch(fptr + KSTAGE, 0, 0);

    #pragma unroll 8
    for (int k = 0; k < KSTAGE; k += 4) {
      // Shared B fragment (4x16 f32): B[K][N] = features[c0+N][K]
      v2f bfrag = *(const v2f*)(fptr + k);
      // A fragments (16x4 f32): lanes 0-15 K={k,k+1}, lanes 16-31 K={k+2,k+3}
      v2f a0 = *(const v2f*)(a0ptr + k);
      v2f a1 = *(const v2f*)(a1ptr + k);
      acc0 = __builtin_amdgcn_wmma_f32_16x16x4_f32(
          false, a0, false, bfrag, (short)0, acc0, false, false);
      acc1 = __builtin_amdgcn_wmma_f32_16x16x4_f32(
          false, a1, false, bfrag, (short)0, acc1, false, false);
    }
  }

  // D layout: VGPR i -> M = i (lanes 0-15), M = 8+i (lanes 16-31); N = l
  float* obase = out + (size_t)b * NM * NC + chunk * 128 + wave * 16 + l;
  const float scale = 1.0f / (float)NHW;   // mean over H*W
  #pragma unroll
  for (int i = 0; i < 8; ++i) {
    int m_local = i + half * 8;
    obase[m_local * NC]        = acc0[i] * scale;   // M rows 0..15
    obase[(16 + m_local) * NC] = acc1[i] * scale;   // M rows 16..31
  }
}

// ---------------------------------------------------------------------------
extern "C" void kernel_launch(void* const* d_in, const int* in_sizes, int n_in,
                              void* d_out, int out_size, void* d_ws, size_t ws_size,
                              hipStream_t stream) {
  const float* features = (const float*)d_in[0];   // [16,1024,32,32]
  const float* raw_att  = (const float*)d_in[1];   // [16,32,32,32]
  float* out_fm  = (float*)d_out;                  // [16,32,1024]
  float* out_att = (float*)d_out + FM_ELEMS;       // [16,32,32,32]

  // 1) attentions = sigmoid(raw)  (also the second tuple output)
  bap_sigmoid_kernel<<<ATT_ELEMS / 4 / 256, 256, 0, stream>>>(raw_att, out_att);

  // 2) batched GEMM with f32 WMMA, reading the attentions written above
  bap_gemm_wmma_kernel<<<NB * 8, 256, 0, stream>>>(features, out_att, out_fm);
}